// RefineMultiBoxLoss_22582938042835
// MI455X (gfx1250) — compile-verified
//
#include <hip/hip_runtime.h>
#include <hip/hip_bf16.h>

// RefineMultiBoxLoss for gfx1250 (MI455X).
// Pipeline: (1) best prior per GT, (2) best GT per prior, (3) force-match
// override, (4) streaming targets + lse/ce + smoothL1 partials,
// (5) per-batch hard-negative mining via LDS-resident bitwise top-k
//     (65KB LDS per workgroup -- CDNA5's 320KB/WGP makes this trivial),
// (6) final reduction using V_WMMA_F32_16X16X4_F32 (ones-matmul reduce).

#define B_ 32
#define P_ 16320
#define C_ 21
#define G_ 50
#define THRESH_ 0.5f

typedef __attribute__((ext_vector_type(2))) float v2f;
typedef __attribute__((ext_vector_type(8))) float v8f;

__device__ __forceinline__ float iou_cp(float ax0, float ay0, float ax1, float ay1,
                                        float bx0, float by0, float bx1, float by1) {
  float ix0 = fmaxf(ax0, bx0), iy0 = fmaxf(ay0, by0);
  float ix1 = fminf(ax1, bx1), iy1 = fminf(ay1, by1);
  float iw = fmaxf(ix1 - ix0, 0.f), ih = fmaxf(iy1 - iy0, 0.f);
  float inter = iw * ih;
  float aa = (ax1 - ax0) * (ay1 - ay0);
  float ab = (bx1 - bx0) * (by1 - by0);
  return inter / (aa + ab - inter);
}

__device__ __forceinline__ float sl1(float x) {
  float d = fabsf(x);
  return d < 1.f ? 0.5f * d * d : d - 0.5f;
}

// ---- K1: per (b,g) argmax over priors of IoU (first-index tie-break) ----
__global__ void k_best_prior(const float* __restrict__ gt, const float* __restrict__ priors,
                             int* __restrict__ bpi) {
  int b = blockIdx.x / G_;
  int g = blockIdx.x % G_;
  const float* gb = gt + ((size_t)b * G_ + g) * 4;
  float ax0 = gb[0], ay0 = gb[1], ax1 = gb[2], ay1 = gb[3];
  float bestv = -1.f; int besti = 0;
  for (int p = threadIdx.x; p < P_; p += 256) {
    const float* pr = priors + (size_t)p * 4;
    float cx = pr[0], cy = pr[1], w = pr[2], h = pr[3];
    float v = iou_cp(ax0, ay0, ax1, ay1,
                     cx - w * 0.5f, cy - h * 0.5f, cx + w * 0.5f, cy + h * 0.5f);
    if (v > bestv) { bestv = v; besti = p; }   // ascending p => strict > keeps first
  }
  __shared__ float sv[256];
  __shared__ int   si[256];
  sv[threadIdx.x] = bestv; si[threadIdx.x] = besti;
  __syncthreads();
  for (int s = 128; s > 0; s >>= 1) {
    if (threadIdx.x < (unsigned)s) {
      float v2 = sv[threadIdx.x + s]; int i2 = si[threadIdx.x + s];
      if (v2 > sv[threadIdx.x] || (v2 == sv[threadIdx.x] && i2 < si[threadIdx.x])) {
        sv[threadIdx.x] = v2; si[threadIdx.x] = i2;
      }
    }
    __syncthreads();
  }
  if (threadIdx.x == 0) bpi[blockIdx.x] = si[0];
}

// ---- K2: per (b,p) max/argmax over GTs (first-index tie-break) ----
__global__ void k_best_truth(const float* __restrict__ gt, const float* __restrict__ priors,
                             float* __restrict__ bto, int* __restrict__ bti) {
  int b = blockIdx.x >> 6;
  int p = ((blockIdx.x & 63) << 8) + threadIdx.x;
  __shared__ float sg[G_ * 4];
  if (threadIdx.x < G_ * 4) sg[threadIdx.x] = gt[(size_t)b * G_ * 4 + threadIdx.x];
  __syncthreads();
  if (p >= P_) return;
  const float* pr = priors + (size_t)p * 4;
  float cx = pr[0], cy = pr[1], w = pr[2], h = pr[3];
  float bx0 = cx - w * 0.5f, by0 = cy - h * 0.5f;
  float bx1 = cx + w * 0.5f, by1 = cy + h * 0.5f;
  float bestv = -1.f; int besti = 0;
  for (int g = 0; g < G_; ++g) {
    float v = iou_cp(sg[g * 4 + 0], sg[g * 4 + 1], sg[g * 4 + 2], sg[g * 4 + 3],
                     bx0, by0, bx1, by1);
    if (v > bestv) { bestv = v; besti = g; }
  }
  size_t bp = (size_t)b * P_ + p;
  bto[bp] = bestv;
  bti[bp] = besti;
}

// ---- K3: force every GT to keep its best prior ----
__global__ void k_override(const int* __restrict__ bpi, float* __restrict__ bto,
                           int* __restrict__ bti) {
  int b = blockIdx.x;
  if (threadIdx.x == 0) {
    for (int g = 0; g < G_; ++g) {
      int p = bpi[b * G_ + g];
      bto[(size_t)b * P_ + p] = 2.0f;
      bti[(size_t)b * P_ + p] = g;
    }
  }
}

// ---- K4: targets + lse/ce + smooth-L1 block partials (streaming pass) ----
__global__ void k_targets(const float* __restrict__ pred_loc, const float* __restrict__ pred_score,
                          const float* __restrict__ priors, const float* __restrict__ gt_boxes,
                          const int* __restrict__ gt_labels,
                          const float* __restrict__ bto, const int* __restrict__ bti,
                          int* __restrict__ confA, float* __restrict__ ceA,
                          float* __restrict__ partial_l) {
  int b = blockIdx.x >> 6;
  int p = ((blockIdx.x & 63) << 8) + threadIdx.x;
  float ll = 0.f;
  if (p < P_) {
    size_t bp = (size_t)b * P_ + p;
    int ti = bti[bp];
    float ov = bto[bp];
    int lbl = gt_labels[b * G_ + ti];
    int cf = (ov < THRESH_) ? 0 : lbl;
    confA[bp] = cf;

    const float* row = pred_score + bp * C_;
    __builtin_prefetch(row + 256 * C_, 0, 0);   // global_prefetch_b8
    float m = row[0];
#pragma unroll
    for (int c = 1; c < C_; ++c) m = fmaxf(m, row[c]);
    float sum = 0.f;
#pragma unroll
    for (int c = 0; c < C_; ++c) sum += expf(row[c] - m);
    float lse = m + logf(sum);
    float ce = lse - row[cf];                   // == CE and == neg mining rank
    ceA[bp] = ce;

    if (cf > 0) {
      const float* gb = gt_boxes + ((size_t)b * G_ + ti) * 4;
      const float* pr = priors + (size_t)p * 4;
      float pcx = pr[0], pcy = pr[1], pw = pr[2], ph = pr[3];
      float tx = ((gb[0] + gb[2]) * 0.5f - pcx) / (0.1f * pw);
      float ty = ((gb[1] + gb[3]) * 0.5f - pcy) / (0.1f * ph);
      float tw = logf((gb[2] - gb[0]) / pw) * 5.0f;   // /VAR1(0.2)
      float th = logf((gb[3] - gb[1]) / ph) * 5.0f;
      const float* pl = pred_loc + bp * 4;
      ll = sl1(pl[0] - tx) + sl1(pl[1] - ty) + sl1(pl[2] - tw) + sl1(pl[3] - th);
    }
  }
  // deterministic block reduce of ll
  __shared__ float wp[8];
  float v = ll;
  for (int o = 16; o > 0; o >>= 1) v += __shfl_down(v, o);
  if ((threadIdx.x & 31) == 0) wp[threadIdx.x >> 5] = v;
  __syncthreads();
  if (threadIdx.x == 0) {
    float s = 0.f;
    for (int i = 0; i < 8; ++i) s += wp[i];
    partial_l[blockIdx.x] = s;
  }
}

// ---- K5: per-batch hard-negative mining; LDS-resident bitwise top-k ----
__global__ __launch_bounds__(1024) void k_mine(const int* __restrict__ confA,
                                               const float* __restrict__ ceA,
                                               float* __restrict__ loss_c_part,
                                               int* __restrict__ numpos_part) {
  __shared__ float r_s[P_];     // 65280 B: rank values (0 for positives)
  __shared__ float wp[32];
  __shared__ int s_cnt;
  __shared__ int s_np;
  int b = blockIdx.x, tid = threadIdx.x;
  size_t base = (size_t)b * P_;

  float posce = 0.f; int poscnt = 0;
  for (int p = tid; p < P_; p += 1024) {
    int cf = confA[base + p];
    float ce = ceA[base + p];
    if (cf > 0) { posce += ce; ++poscnt; r_s[p] = 0.f; }
    else        { r_s[p] = ce; }                         // ce >= 0 always
  }
  if (tid == 0) s_np = 0;
  __syncthreads();
  {
    int v = poscnt;
    for (int o = 16; o > 0; o >>= 1) v += __shfl_down(v, o);
    if ((tid & 31) == 0) atomicAdd(&s_np, v);
  }
  float pv = posce;
  for (int o = 16; o > 0; o >>= 1) pv += __shfl_down(pv, o);
  if ((tid & 31) == 0) wp[tid >> 5] = pv;
  __syncthreads();
  int num_pos = s_np;
  int num_neg = min(3 * num_pos, P_ - 1);
  float posce_tot = 0.f;
  if (tid == 0) { for (int i = 0; i < 32; ++i) posce_tot += wp[i]; }

  // bitwise binary search for k-th largest value (bits monotone: r >= 0)
  unsigned t = 0u;
  if (num_neg > 0) {
    for (int bit = 31; bit >= 0; --bit) {
      unsigned cand = t | (1u << bit);
      __syncthreads();
      if (tid == 0) s_cnt = 0;
      __syncthreads();
      int c = 0;
      for (int p = tid; p < P_; p += 1024)
        if (__float_as_uint(r_s[p]) >= cand) ++c;
      for (int o = 16; o > 0; o >>= 1) c += __shfl_down(c, o);
      if ((tid & 31) == 0) atomicAdd(&s_cnt, c);
      __syncthreads();
      if (s_cnt >= num_neg) t = cand;   // uniform across block
    }
  }
  // sum of strictly-greater elements + their count
  __syncthreads();
  if (tid == 0) s_cnt = 0;
  __syncthreads();
  int cgt = 0; float sgt = 0.f;
  for (int p = tid; p < P_; p += 1024) {
    float r = r_s[p];
    if (__float_as_uint(r) > t) { ++cgt; sgt += r; }
  }
  for (int o = 16; o > 0; o >>= 1) { cgt += __shfl_down(cgt, o); sgt += __shfl_down(sgt, o); }
  if ((tid & 31) == 0) { atomicAdd(&s_cnt, cgt); wp[tid >> 5] = sgt; }
  __syncthreads();
  if (tid == 0) {
    float lc = posce_tot;
    if (num_neg > 0) {
      float sum_gt = 0.f;
      for (int i = 0; i < 32; ++i) sum_gt += wp[i];
      int need = num_neg - s_cnt;                 // ties all equal value(t)
      if (need < 0) need = 0;
      lc += sum_gt + (float)need * __uint_as_float(t);
    }
    loss_c_part[b] = lc;
    numpos_part[b] = num_pos;
  }
}

// ---- K6: final combine. WMMA ones-matmul reduction of 2048 partials ----
// A (16x4 f32): lanes 0-15 hold (M=lane, K=0..1), lanes 16-31 hold (M=lane-16, K=2..3).
// B = ones -> D[m][n] = rowsum(m); accumulate C across 32 chunks of 64.
__global__ void k_final(const float* __restrict__ partial_l,
                        const float* __restrict__ loss_c_part,
                        const int* __restrict__ numpos_part,
                        float* __restrict__ out) {
  int lane = threadIdx.x;              // 32 threads = one full wave, EXEC all-1s
  int m = lane & 15;
  int koff = (lane >> 4) << 1;         // 0 or 2
  v8f acc = {0.f, 0.f, 0.f, 0.f, 0.f, 0.f, 0.f, 0.f};
  v2f ones; ones.x = 1.f; ones.y = 1.f;
  for (int i = 0; i < 32; ++i) {
    const float* bp = partial_l + i * 64 + m * 4 + koff;
    v2f a; a.x = bp[0]; a.y = bp[1];
    acc = __builtin_amdgcn_wmma_f32_16x16x4_f32(
        /*neg_a=*/false, a, /*neg_b=*/false, ones,
        /*c_mod=*/(short)0, acc, /*reuse_a=*/false, /*reuse_b=*/false);
  }
  float s = acc[0] + acc[1] + acc[2] + acc[3] + acc[4] + acc[5] + acc[6] + acc[7];
  float total_l = s + __shfl_down(s, 16);     // lane0: rows 0-7 + rows 8-15

  float lc = loss_c_part[lane];
  int np = numpos_part[lane];
  for (int o = 16; o > 0; o >>= 1) { lc += __shfl_down(lc, o); np += __shfl_down(np, o); }

  if (lane == 0) {
    float N = (float)np;
    out[0] = total_l / N;
    out[1] = lc / N;
  }
}

extern "C" void kernel_launch(void* const* d_in, const int* in_sizes, int n_in,
                              void* d_out, int out_size, void* d_ws, size_t ws_size,
                              hipStream_t stream) {
  const float* pred_loc   = (const float*)d_in[0];
  const float* pred_score = (const float*)d_in[1];
  const float* priors     = (const float*)d_in[2];
  const float* gt_boxes   = (const float*)d_in[3];
  const int*   gt_labels  = (const int*)d_in[4];

  char* w = (char*)d_ws;
  size_t off = 0;
  auto alloc = [&](size_t bytes) -> void* {
    void* p = w + off;
    off = (off + bytes + 255) & ~(size_t)255;
    return p;
  };
  float* bto       = (float*)alloc(sizeof(float) * (size_t)B_ * P_);
  int*   bti       = (int*)  alloc(sizeof(int)   * (size_t)B_ * P_);
  int*   confA     = (int*)  alloc(sizeof(int)   * (size_t)B_ * P_);
  float* ceA       = (float*)alloc(sizeof(float) * (size_t)B_ * P_);
  int*   bpi       = (int*)  alloc(sizeof(int)   * (size_t)B_ * G_);
  float* partial_l = (float*)alloc(sizeof(float) * 2048);
  float* lcp       = (float*)alloc(sizeof(float) * B_);
  int*   npp       = (int*)  alloc(sizeof(int)   * B_);

  hipLaunchKernelGGL(k_best_prior, dim3(B_ * G_), dim3(256), 0, stream, gt_boxes, priors, bpi);
  hipLaunchKernelGGL(k_best_truth, dim3(B_ * 64), dim3(256), 0, stream, gt_boxes, priors, bto, bti);
  hipLaunchKernelGGL(k_override,   dim3(B_),      dim3(32),  0, stream, bpi, bto, bti);
  hipLaunchKernelGGL(k_targets,    dim3(B_ * 64), dim3(256), 0, stream,
                     pred_loc, pred_score, priors, gt_boxes, gt_labels,
                     bto, bti, confA, ceA, partial_l);
  hipLaunchKernelGGL(k_mine,       dim3(B_),      dim3(1024), 0, stream, confA, ceA, lcp, npp);
  hipLaunchKernelGGL(k_final,      dim3(1),       dim3(32),  0, stream, partial_l, lcp, npp,
                     (float*)d_out);
}